// MultilabelCenterTrimmedLoss_88691074662476
// MI455X (gfx1250) — compile-verified
//
#include <hip/hip_runtime.h>
#include <stdint.h>
#include <math.h>

// ---------------------------------------------------------------------------
// MultilabelCenterTrimmedLoss on MI455X (gfx1250).
// Memory-bound (~345MB traffic -> ~15us @ 23.3TB/s). Exact per-sample top-k
// via 3-level radix select on the non-negative neg-focal-loss key bits.
// All streaming loops use b128 loads; wave reductions run on the matrix pipe
// via V_WMMA_F32_16X16X4_F32 (A=data, B=ones => row sums, two-stage fold).
// ---------------------------------------------------------------------------

#define NB    8
#define HH    768
#define WW    768
#define HWN   (HH * WW)       // 589824
#define N4    (HWN / 4)       // 147456 float4 groups per sample
#define KTRIM (HWN / 16)      // 36864
#define IGN   (-100.0f)

// workspace layout (32-bit words)
#define WS_ACC    0                    // [0]=pos [1]=neg [2]=vessel [3]=fishing [4]=offset [5]=size (f32)
#define WS_NOBJ   6                    // f32 count of peaks
#define WS_NVES   7                    // f32 count of vessel&peaks
#define WS_QCNT   8                    // u32 x8  per-sample tie-quota counters
#define WS_TH     16                   // u32 x8  per-sample threshold key
#define WS_QUOTA  24                   // u32 x8  per-sample #ties to mask
#define WS_BIN1   32                   // u32 x8
#define WS_PFX1   40                   // u32 x8
#define WS_BIN2   48                   // u32 x8
#define WS_PFX2   56                   // u32 x8
#define WS_BIN3   64                   // u32 x8
#define WS_PFX3   72                   // u32 x8
#define WS_H1     80                   // u32 8*2048
#define WS_H2     (WS_H1 + NB * 2048)  // u32 8*2048
#define WS_H3     (WS_H2 + NB * 2048)  // u32 8*1024
#define WS_TOTAL  (WS_H3 + NB * 1024)  // 41040 words (~164 KB, 16B-aligned)
#define WS_KEYS   WS_TOTAL             // optional u32 8*HWN key cache (18.9 MB)

typedef float v2f __attribute__((ext_vector_type(2)));
typedef float v8f __attribute__((ext_vector_type(8)));

// ---------------- math helpers ----------------------------------------------

__device__ __forceinline__ float logsig(float x) {
  return fminf(x, 0.0f) - log1pf(expf(-fabsf(x)));
}
__device__ __forceinline__ float sigm(float x) {
  return 1.0f / (1.0f + expf(-x));
}
__device__ __forceinline__ float entropy_logits(float x) {
  float p = sigm(x);
  return -(p * logsig(x) + (1.0f - p) * logsig(-x));
}
__device__ __forceinline__ float bce_logits(float x, float t) {
  return -(t * logsig(x) + (1.0f - t) * logsig(-x));
}
// reduced focal negative term; >= 0 always (IEEE bits order-monotonic)
__device__ __forceinline__ float neg_loss(float gt, float po) {
  if (gt == IGN || gt == 1.0f) return 0.0f;
  float pt = sigm(po);
  float om = 1.0f - gt;
  float w  = om * om;
  w = w * w;                                  // (1-gt)^4
  return -w * pt * pt * logsig(-po);
}
__device__ __forceinline__ float pos_term(float po) {
  float om = 1.0f - sigm(po);
  return -(om * om) * logsig(po);
}

// ---------------- wave32 sum-reduction on the matrix pipe --------------------
__device__ __forceinline__ float wave_sum_wmma(float v) {
  v2f ones; ones.x = 1.0f; ones.y = 1.0f;
  v2f a;    a.x = v;       a.y = 0.0f;
  v8f c = {};
  v8f d = __builtin_amdgcn_wmma_f32_16x16x4_f32(false, a, false, ones,
                                                (short)0, c, false, false);
  float s = d[0] + d[1] + d[2] + d[3] + d[4] + d[5] + d[6] + d[7];
  v2f a2;   a2.x = s;      a2.y = 0.0f;
  v8f d2 = __builtin_amdgcn_wmma_f32_16x16x4_f32(false, a2, false, ones,
                                                 (short)0, c, false, false);
  return d2[0];
}

// ---------------- block-parallel radix pick ---------------------------------
// Finds bin b (descending) with prefix(count of bins>b) + h[b] >= target,
// prefix < target. 256 threads: per-chunk sums -> tiny serial scan in LDS.
template <int NBINS>
__device__ __forceinline__ void block_radix_pick(const uint32_t* __restrict__ h,
                                                 uint32_t target,
                                                 uint32_t* outBin,
                                                 uint32_t* outPrefix) {
  __shared__ uint32_t csum[256];
  constexpr int PER = NBINS / 256;
  const int t = threadIdx.x;
  uint32_t sum = 0;
  for (int j = 0; j < PER; ++j) sum += h[t * PER + j];
  csum[t] = sum;
  __syncthreads();
  if (t == 0) {
    uint32_t prefix = 0;
    int c = 255;
    for (; c > 0; --c) {
      uint32_t cs = csum[c];
      if (prefix + cs >= target) break;
      prefix += cs;
    }
    uint32_t bin = (uint32_t)(c * PER);
    for (int b = PER - 1; b >= 0; --b) {
      uint32_t cc = h[c * PER + b];
      if (prefix + cc >= target) { bin = (uint32_t)(c * PER + b); break; }
      prefix += cc;
    }
    *outBin = bin;
    *outPrefix = prefix;
  }
}

// ---------------- kernels ----------------------------------------------------

__global__ void k_init(uint32_t* ws) {
  int i = blockIdx.x * blockDim.x + threadIdx.x;
  if (i < WS_TOTAL) ws[i] = 0u;
}

// pass 1: level-1 histogram (key bits [31:21]) + peak/vessel counts + pos loss
// optionally stores key bits to the cache.
__global__ void __launch_bounds__(256) k_hist1(const float* __restrict__ gt_obj,
                                               const float* __restrict__ p_obj,
                                               const float* __restrict__ gt_ves,
                                               uint32_t* __restrict__ keys,
                                               uint32_t* __restrict__ ws) {
  __shared__ uint32_t h[2048];
  for (int i = threadIdx.x; i < 2048; i += blockDim.x) h[i] = 0u;
  __syncthreads();

  const int s = blockIdx.y;
  const size_t base = (size_t)s * HWN;
  const float4* g4 = (const float4*)(gt_obj + base);
  const float4* p4 = (const float4*)(p_obj + base);
  const float4* v4 = (const float4*)(gt_ves + base);
  uint4* k4 = keys ? ((uint4*)keys) + (size_t)s * N4 : nullptr;

  float posAcc = 0.0f, objAcc = 0.0f, vesAcc = 0.0f;

  for (int i = blockIdx.x * blockDim.x + threadIdx.x; i < N4;
       i += gridDim.x * blockDim.x) {
    float4 g = g4[i];
    float4 p = p4[i];
    float ga[4] = {g.x, g.y, g.z, g.w};
    float pa[4] = {p.x, p.y, p.z, p.w};
    uint4 kk;
    uint32_t* ka = (uint32_t*)&kk;
#pragma unroll
    for (int e = 0; e < 4; ++e) {
      ka[e] = __float_as_uint(neg_loss(ga[e], pa[e]));
      atomicAdd(&h[ka[e] >> 21], 1u);
    }
    if (k4) k4[i] = kk;

    bool pk0 = (ga[0] == 1.0f), pk1 = (ga[1] == 1.0f);
    bool pk2 = (ga[2] == 1.0f), pk3 = (ga[3] == 1.0f);
    if (pk0 | pk1 | pk2 | pk3) {
      float4 v = v4[i];
      float va[4] = {v.x, v.y, v.z, v.w};
      bool pk[4] = {pk0, pk1, pk2, pk3};
#pragma unroll
      for (int e = 0; e < 4; ++e) {
        if (pk[e]) {
          objAcc += 1.0f;
          if (va[e] == 1.0f) vesAcc += 1.0f;
          posAcc += pos_term(pa[e]);
        }
      }
    }
  }
  __syncthreads();
  for (int i = threadIdx.x; i < 2048; i += blockDim.x)
    if (h[i]) atomicAdd(&ws[WS_H1 + s * 2048 + i], h[i]);

  float a0 = wave_sum_wmma(posAcc);
  float a1 = wave_sum_wmma(objAcc);
  float a2 = wave_sum_wmma(vesAcc);
  if ((threadIdx.x & 31) == 0) {
    float* wf = (float*)ws;
    atomicAdd(&wf[WS_ACC + 0], a0);
    atomicAdd(&wf[WS_NOBJ], a1);
    atomicAdd(&wf[WS_NVES], a2);
  }
}

__global__ void __launch_bounds__(256) k_select1(uint32_t* ws) {
  int s = blockIdx.x;
  block_radix_pick<2048>(&ws[WS_H1 + s * 2048], (uint32_t)KTRIM,
                         &ws[WS_BIN1 + s], &ws[WS_PFX1 + s]);
}

// pass 2: level-2 histogram (key bits [20:10]) restricted to bin1
__global__ void __launch_bounds__(256) k_hist2(const float* __restrict__ gt_obj,
                                               const float* __restrict__ p_obj,
                                               const uint32_t* __restrict__ keys,
                                               uint32_t* __restrict__ ws) {
  __shared__ uint32_t h[2048];
  for (int i = threadIdx.x; i < 2048; i += blockDim.x) h[i] = 0u;
  __syncthreads();
  const int s = blockIdx.y;
  const uint32_t b1 = ws[WS_BIN1 + s];
  const size_t base = (size_t)s * HWN;
  const float4* g4 = (const float4*)(gt_obj + base);
  const float4* p4 = (const float4*)(p_obj + base);
  const uint4* k4 = keys ? ((const uint4*)keys) + (size_t)s * N4 : nullptr;

  for (int i = blockIdx.x * blockDim.x + threadIdx.x; i < N4;
       i += gridDim.x * blockDim.x) {
    uint4 kk;
    if (k4) {
      kk = k4[i];
    } else {
      float4 g = g4[i], p = p4[i];
      kk.x = __float_as_uint(neg_loss(g.x, p.x));
      kk.y = __float_as_uint(neg_loss(g.y, p.y));
      kk.z = __float_as_uint(neg_loss(g.z, p.z));
      kk.w = __float_as_uint(neg_loss(g.w, p.w));
    }
    uint32_t* ka = (uint32_t*)&kk;
#pragma unroll
    for (int e = 0; e < 4; ++e)
      if ((ka[e] >> 21) == b1) atomicAdd(&h[(ka[e] >> 10) & 0x7FFu], 1u);
  }
  __syncthreads();
  for (int i = threadIdx.x; i < 2048; i += blockDim.x)
    if (h[i]) atomicAdd(&ws[WS_H2 + s * 2048 + i], h[i]);
}

__global__ void __launch_bounds__(256) k_select2(uint32_t* ws) {
  int s = blockIdx.x;
  uint32_t k2 = (uint32_t)KTRIM - ws[WS_PFX1 + s];
  block_radix_pick<2048>(&ws[WS_H2 + s * 2048], k2,
                         &ws[WS_BIN2 + s], &ws[WS_PFX2 + s]);
}

// pass 3: level-3 histogram (key bits [9:0]) restricted to (bin1,bin2)
__global__ void __launch_bounds__(256) k_hist3(const float* __restrict__ gt_obj,
                                               const float* __restrict__ p_obj,
                                               const uint32_t* __restrict__ keys,
                                               uint32_t* __restrict__ ws) {
  __shared__ uint32_t h[1024];
  for (int i = threadIdx.x; i < 1024; i += blockDim.x) h[i] = 0u;
  __syncthreads();
  const int s = blockIdx.y;
  const uint32_t hi = (ws[WS_BIN1 + s] << 11) | ws[WS_BIN2 + s];
  const size_t base = (size_t)s * HWN;
  const float4* g4 = (const float4*)(gt_obj + base);
  const float4* p4 = (const float4*)(p_obj + base);
  const uint4* k4 = keys ? ((const uint4*)keys) + (size_t)s * N4 : nullptr;

  for (int i = blockIdx.x * blockDim.x + threadIdx.x; i < N4;
       i += gridDim.x * blockDim.x) {
    uint4 kk;
    if (k4) {
      kk = k4[i];
    } else {
      float4 g = g4[i], p = p4[i];
      kk.x = __float_as_uint(neg_loss(g.x, p.x));
      kk.y = __float_as_uint(neg_loss(g.y, p.y));
      kk.z = __float_as_uint(neg_loss(g.z, p.z));
      kk.w = __float_as_uint(neg_loss(g.w, p.w));
    }
    uint32_t* ka = (uint32_t*)&kk;
#pragma unroll
    for (int e = 0; e < 4; ++e)
      if ((ka[e] >> 10) == hi) atomicAdd(&h[ka[e] & 0x3FFu], 1u);
  }
  __syncthreads();
  for (int i = threadIdx.x; i < 1024; i += blockDim.x)
    if (h[i]) atomicAdd(&ws[WS_H3 + s * 1024 + i], h[i]);
}

__global__ void __launch_bounds__(256) k_select3(uint32_t* ws) {
  int s = blockIdx.x;
  uint32_t k3 = (uint32_t)KTRIM - ws[WS_PFX1 + s] - ws[WS_PFX2 + s];
  block_radix_pick<1024>(&ws[WS_H3 + s * 1024], k3,
                         &ws[WS_BIN3 + s], &ws[WS_PFX3 + s]);
  if (threadIdx.x == 0) {
    uint32_t b3 = ws[WS_BIN3 + s];
    uint32_t p3 = ws[WS_PFX3 + s];
    ws[WS_TH + s]    = (ws[WS_BIN1 + s] << 21) | (ws[WS_BIN2 + s] << 10) | b3;
    ws[WS_QUOTA + s] = k3 - p3;  // #elements equal to T that get masked
    ws[WS_QCNT + s]  = 0u;
  }
}

// pass 4: the five non-pos loss sums with trimming mask applied
__global__ void __launch_bounds__(256) k_final(
    const float* __restrict__ gt_obj, const float* __restrict__ gt_ves,
    const float* __restrict__ gt_fis, const float* __restrict__ gt_size,
    const float* __restrict__ gt_off, const float* __restrict__ p_obj,
    const float* __restrict__ p_ves, const float* __restrict__ p_fis,
    const float* __restrict__ p_size, const float* __restrict__ p_off,
    const uint32_t* __restrict__ keys, uint32_t* __restrict__ ws) {
  const int s = blockIdx.y;
  const uint32_t T = ws[WS_TH + s];
  const uint32_t quota = ws[WS_QUOTA + s];
  const size_t base  = (size_t)s * HWN;
  const size_t base2 = (size_t)s * 2 * HWN;  // (B,2,H,W)

  const float4* g4  = (const float4*)(gt_obj + base);
  const float4* p4  = (const float4*)(p_obj + base);
  const float4* gv4 = (const float4*)(gt_ves + base);
  const float4* gf4 = (const float4*)(gt_fis + base);
  const float4* pv4 = (const float4*)(p_ves + base);
  const float4* pf4 = (const float4*)(p_fis + base);
  const uint4*  k4  = keys ? ((const uint4*)keys) + (size_t)s * N4 : nullptr;

  float negA = 0.0f, vesA = 0.0f, fisA = 0.0f, offA = 0.0f, sizA = 0.0f;

  for (int i = blockIdx.x * blockDim.x + threadIdx.x; i < N4;
       i += gridDim.x * blockDim.x) {
    float4 g = g4[i], p = p4[i];
    float4 gv = gv4[i], gf = gf4[i], pv = pv4[i], pf = pf4[i];
    float ga[4]  = {g.x, g.y, g.z, g.w};
    float pa[4]  = {p.x, p.y, p.z, p.w};
    float gva[4] = {gv.x, gv.y, gv.z, gv.w};
    float gfa[4] = {gf.x, gf.y, gf.z, gf.w};
    float pva[4] = {pv.x, pv.y, pv.z, pv.w};
    float pfa[4] = {pf.x, pf.y, pf.z, pf.w};
    uint4 kk;
    if (k4) {
      kk = k4[i];
    } else {
      kk.x = __float_as_uint(neg_loss(ga[0], pa[0]));
      kk.y = __float_as_uint(neg_loss(ga[1], pa[1]));
      kk.z = __float_as_uint(neg_loss(ga[2], pa[2]));
      kk.w = __float_as_uint(neg_loss(ga[3], pa[3]));
    }
    uint32_t* ka = (uint32_t*)&kk;

#pragma unroll
    for (int e = 0; e < 4; ++e) {
      float gt = ga[e], po = pa[e];
      uint32_t key = ka[e];
      bool masked = key > T;
      if (key == T) {
        uint32_t r = atomicAdd(&ws[WS_QCNT + s], 1u);
        masked = (r < quota);
      }
      if (masked) {
        negA += entropy_logits(po);
        vesA += entropy_logits(pva[e]);
        fisA += entropy_logits(pfa[e]);
      } else {
        negA += __uint_as_float(key);
        float cert_w = (gt == IGN) ? 0.0f : gt;  // gt_objectness * cert_obj
        if (gva[e] != IGN) vesA += bce_logits(pva[e], gva[e]) * cert_w;
        if (gfa[e] != IGN) fisA += bce_logits(pfa[e], gfa[e]) * cert_w;
        if (gt == 1.0f) {  // peaks: size + offset L2 terms (rare -> scalar)
          int idx = i * 4 + e;
          float ds = p_size[base + idx] - gt_size[base + idx];
          sizA += ds * ds;
          float d0 = p_off[base2 + idx]       - gt_off[base2 + idx];
          float d1 = p_off[base2 + HWN + idx] - gt_off[base2 + HWN + idx];
          offA += d0 * d0 + d1 * d1;
        }
      }
    }
  }

  negA = wave_sum_wmma(negA);
  vesA = wave_sum_wmma(vesA);
  fisA = wave_sum_wmma(fisA);
  offA = wave_sum_wmma(offA);
  sizA = wave_sum_wmma(sizA);
  if ((threadIdx.x & 31) == 0) {
    float* wf = (float*)ws;
    atomicAdd(&wf[WS_ACC + 1], negA);
    atomicAdd(&wf[WS_ACC + 2], vesA);
    atomicAdd(&wf[WS_ACC + 3], fisA);
    atomicAdd(&wf[WS_ACC + 4], offA);
    atomicAdd(&wf[WS_ACC + 5], sizA);
  }
}

__global__ void k_finalize(const uint32_t* __restrict__ ws,
                           float* __restrict__ out) {
  if (threadIdx.x == 0 && blockIdx.x == 0) {
    const float* wf = (const float*)ws;
    float no = fmaxf(wf[WS_NOBJ], 1.0f);
    float nv = fmaxf(wf[WS_NVES], 1.0f);
    out[0] = wf[WS_ACC + 0] / no;  // pos
    out[1] = wf[WS_ACC + 1] / no;  // neg (trimmed)
    out[2] = wf[WS_ACC + 2] / nv;  // vessel
    out[3] = wf[WS_ACC + 3] / nv;  // fishing
    out[4] = wf[WS_ACC + 4] / no;  // offset
    out[5] = wf[WS_ACC + 5] / no;  // size
  }
}

// ---------------------------------------------------------------------------

extern "C" void kernel_launch(void* const* d_in, const int* in_sizes, int n_in,
                              void* d_out, int out_size, void* d_ws,
                              size_t ws_size, hipStream_t stream) {
  (void)in_sizes; (void)n_in; (void)out_size;
  const float* gt_obj  = (const float*)d_in[0];
  const float* gt_ves  = (const float*)d_in[1];
  const float* gt_fis  = (const float*)d_in[2];
  const float* gt_size = (const float*)d_in[3];
  const float* gt_off  = (const float*)d_in[4];
  const float* p_obj   = (const float*)d_in[5];
  const float* p_ves   = (const float*)d_in[6];
  const float* p_fis   = (const float*)d_in[7];
  const float* p_size  = (const float*)d_in[8];
  const float* p_off   = (const float*)d_in[9];
  uint32_t* ws = (uint32_t*)d_ws;
  float* out = (float*)d_out;

  // host-side (deterministic) choice: cache neg-loss keys if workspace fits
  const size_t need_cached = (size_t)(WS_TOTAL + (size_t)NB * HWN) * 4u;
  uint32_t* keys = (ws_size >= need_cached) ? (ws + WS_KEYS) : nullptr;

  const dim3 blk(256);
  const dim3 gpass(96, NB);  // 96*256 threads/sample, 6 float4-iters each

  k_init<<<dim3((WS_TOTAL + 255) / 256), blk, 0, stream>>>(ws);
  k_hist1<<<gpass, blk, 0, stream>>>(gt_obj, p_obj, gt_ves, keys, ws);
  k_select1<<<dim3(NB), blk, 0, stream>>>(ws);
  k_hist2<<<gpass, blk, 0, stream>>>(gt_obj, p_obj, keys, ws);
  k_select2<<<dim3(NB), blk, 0, stream>>>(ws);
  k_hist3<<<gpass, blk, 0, stream>>>(gt_obj, p_obj, keys, ws);
  k_select3<<<dim3(NB), blk, 0, stream>>>(ws);
  k_final<<<gpass, blk, 0, stream>>>(gt_obj, gt_ves, gt_fis, gt_size, gt_off,
                                     p_obj, p_ves, p_fis, p_size, p_off,
                                     keys, ws);
  k_finalize<<<dim3(1), dim3(32), 0, stream>>>(ws, out);
}